// GATConv_55490977464719
// MI455X (gfx1250) — compile-verified
//
#include <hip/hip_runtime.h>
#include <hip/hip_bf16.h>
#include <climits>

typedef float v2f __attribute__((ext_vector_type(2)));
typedef float v8f __attribute__((ext_vector_type(8)));

// ---------------------------------------------------------------------------
// K0: out = bias broadcast; alpha_max = INT_MIN (encoded); alpha_sum = 0
// ---------------------------------------------------------------------------
__global__ __launch_bounds__(256) void gat_init(float* __restrict__ out,
                                                const float* __restrict__ bias,
                                                int out_elems,
                                                int* __restrict__ amax,
                                                float* __restrict__ asum,
                                                int nh) {
  int i = blockIdx.x * blockDim.x + threadIdx.x;
  if (i < out_elems) out[i] = bias[i & 255];
  if (i < nh) { amax[i] = INT_MIN; asum[i] = 0.0f; }
}

// ---------------------------------------------------------------------------
// K1: x_h[N,256] = x[N,256] @ W'[256,256] with W'[k][h*64+c] = weight[h][k][c]
// fp32 WMMA 16x16x4; weights staged through LDS in 64KB chunks.
// Block = 256 threads (8 waves), one 16-row tile per block.
// Wave w owns column tiles {w, w+8}.
// ---------------------------------------------------------------------------
__global__ __launch_bounds__(256) void gat_gemm(const float* __restrict__ x,
                                                const float* __restrict__ weight,
                                                float* __restrict__ xh,
                                                int n_nodes) {
  __shared__ float Bs[64 * 256];  // 64 KB chunk of W' (K x N)

  const int rt       = blockIdx.x;       // row tile
  const int tid      = threadIdx.x;
  const int lane     = tid & 31;
  const int wave     = tid >> 5;         // 0..7
  const int halfLane = lane & 15;
  const int hiHalf   = lane >> 4;        // 0 for lanes 0-15, 1 for 16-31

  const int ct0 = wave;
  const int ct1 = wave + 8;

  const int rowBase = rt * 16;
  int aRowIdx = rowBase + halfLane;
  if (aRowIdx >= n_nodes) aRowIdx = n_nodes - 1;       // clamp (loads only)
  const float* aRow = x + (size_t)aRowIdx * 256;

  v8f acc0 = {};
  v8f acc1 = {};

  for (int kc = 0; kc < 256; kc += 64) {
    // Stage Bs[kk][n] = W'[kc+kk][n],  n = h*64+c -> weight[h*16384 + k*64 + c]
    for (int i = tid; i < 64 * 256; i += 256) {
      int kk = i >> 8;
      int n  = i & 255;
      int h  = n >> 6;
      int c  = n & 63;
      Bs[i] = weight[h * 16384 + (kc + kk) * 64 + c];
    }
    __syncthreads();

    #pragma unroll 4
    for (int k0 = 0; k0 < 64; k0 += 4) {
      // A operand (16x4 fp32): lanes 0-15 hold k0,k0+1; lanes 16-31 hold k0+2,k0+3
      const int kA = kc + k0 + hiHalf * 2;
      v2f a;
      a.x = aRow[kA];
      a.y = aRow[kA + 1];

      // B operand (4x16 fp32), mirrored layout, from LDS
      const int kB = k0 + hiHalf * 2;
      v2f b0, b1;
      b0.x = Bs[(kB + 0) * 256 + ct0 * 16 + halfLane];
      b0.y = Bs[(kB + 1) * 256 + ct0 * 16 + halfLane];
      b1.x = Bs[(kB + 0) * 256 + ct1 * 16 + halfLane];
      b1.y = Bs[(kB + 1) * 256 + ct1 * 16 + halfLane];

      acc0 = __builtin_amdgcn_wmma_f32_16x16x4_f32(false, a, false, b0,
                                                   (short)0, acc0, false, false);
      acc1 = __builtin_amdgcn_wmma_f32_16x16x4_f32(false, a, false, b1,
                                                   (short)0, acc1, false, false);
    }
    __syncthreads();
  }

  // C/D layout: VGPR j -> (M=j, lanes 0-15) / (M=j+8, lanes 16-31), N = lane%16
  const int colBase0 = ct0 * 16 + halfLane;
  const int colBase1 = ct1 * 16 + halfLane;
  const int mOff     = hiHalf * 8;
  #pragma unroll
  for (int j = 0; j < 8; ++j) {
    const int row = rowBase + mOff + j;
    if (row < n_nodes) {
      xh[(size_t)row * 256 + colBase0] = acc0[j];
      xh[(size_t)row * 256 + colBase1] = acc1[j];
    }
  }
}

// ---------------------------------------------------------------------------
// K2: alpha_src/alpha_dst = <x_h[n,h,:], att_{src,dst}[h,:]>
// ---------------------------------------------------------------------------
__global__ __launch_bounds__(256) void gat_alpha_node(const float* __restrict__ xh,
                                                      const float* __restrict__ att_src,
                                                      const float* __restrict__ att_dst,
                                                      float* __restrict__ asrc,
                                                      float* __restrict__ adst,
                                                      int n_nodes) {
  int idx = blockIdx.x * blockDim.x + threadIdx.x;  // n*4 + h
  if (idx >= n_nodes * 4) return;
  const int h = idx & 3;
  const int n = idx >> 2;
  const float* v = xh + (size_t)n * 256 + h * 64;
  const float* s = att_src + h * 64;
  const float* d = att_dst + h * 64;
  float ss = 0.0f, dd = 0.0f;
  #pragma unroll 8
  for (int c = 0; c < 64; ++c) {
    const float xv = v[c];
    ss += xv * s[c];
    dd += xv * d[c];
  }
  asrc[idx] = ss;
  adst[idx] = dd;
}

// monotone int encoding for exact float atomicMax
__device__ __forceinline__ int enc_f32(float f) {
  int i = __float_as_int(f);
  return (i >= 0) ? i : (i ^ 0x7FFFFFFF);
}
__device__ __forceinline__ float dec_f32(int i) {
  return __int_as_float((i >= 0) ? i : (i ^ 0x7FFFFFFF));
}

// ---------------------------------------------------------------------------
// K3: alpha = leaky_relu(asrc[row] + adst[col]); segment max by row
// ---------------------------------------------------------------------------
__global__ __launch_bounds__(256) void gat_edge_max(const int* __restrict__ row,
                                                    const int* __restrict__ col,
                                                    const float* __restrict__ asrc,
                                                    const float* __restrict__ adst,
                                                    float* __restrict__ alphaE,
                                                    int* __restrict__ amax,
                                                    int n_edges) {
  int idx = blockIdx.x * blockDim.x + threadIdx.x;  // e*4 + h
  if (idx >= n_edges * 4) return;
  const int h = idx & 3;
  const int e = idx >> 2;
  const int r = row[e];
  const int c = col[e];
  float a = asrc[r * 4 + h] + adst[c * 4 + h];
  a = (a > 0.0f) ? a : 0.2f * a;  // leaky relu, slope 0.2
  alphaE[idx] = a;
  atomicMax(&amax[r * 4 + h], enc_f32(a));
}

// ---------------------------------------------------------------------------
// K4: alpha = exp(alpha - max[row]); segment sum by row
// ---------------------------------------------------------------------------
__global__ __launch_bounds__(256) void gat_edge_exp(const int* __restrict__ row,
                                                    float* __restrict__ alphaE,
                                                    const int* __restrict__ amax,
                                                    float* __restrict__ asum,
                                                    int n_edges) {
  int idx = blockIdx.x * blockDim.x + threadIdx.x;
  if (idx >= n_edges * 4) return;
  const int h = idx & 3;
  const int e = idx >> 2;
  const int r = row[e];
  const float m  = dec_f32(amax[r * 4 + h]);
  const float ex = __expf(alphaE[idx] - m);
  alphaE[idx] = ex;
  atomicAdd(&asum[r * 4 + h], ex);
}

// ---------------------------------------------------------------------------
// K5: out[row] += (alpha / clip(sum[row])) * x_h[col]   (wave per edge-head)
// ---------------------------------------------------------------------------
__global__ __launch_bounds__(256) void gat_scatter(const int* __restrict__ row,
                                                   const int* __restrict__ col,
                                                   const float* __restrict__ alphaE,
                                                   const float* __restrict__ asum,
                                                   const float* __restrict__ xh,
                                                   float* __restrict__ out,
                                                   int n_edges) {
  const int gw   = blockIdx.x * 8 + (threadIdx.x >> 5);  // global wave = e*4+h
  const int lane = threadIdx.x & 31;
  const int e = gw >> 2;
  const int h = gw & 3;
  if (e >= n_edges) return;
  const int r = row[e];
  const int c = col[e];
  float s = asum[r * 4 + h];
  s = (s > 1e-16f) ? s : 1e-16f;
  const float w = alphaE[e * 4 + h] / s;

  const float2 xv = *(const float2*)(xh + (size_t)c * 256 + h * 64 + lane * 2);
  float* op = out + (size_t)r * 256 + h * 64 + lane * 2;
  atomicAdd(op,     w * xv.x);
  atomicAdd(op + 1, w * xv.y);
}

// ---------------------------------------------------------------------------
extern "C" void kernel_launch(void* const* d_in, const int* in_sizes, int n_in,
                              void* d_out, int out_size, void* d_ws, size_t ws_size,
                              hipStream_t stream) {
  const float* x        = (const float*)d_in[0];
  const int*   edge     = (const int*)d_in[1];
  const float* weight   = (const float*)d_in[2];
  const float* att_src  = (const float*)d_in[3];
  const float* att_dst  = (const float*)d_in[4];
  const float* bias     = (const float*)d_in[5];
  float*       out      = (float*)d_out;

  const int N = in_sizes[0] / 256;   // 50000
  const int E = in_sizes[1] / 2;     // 800000

  // workspace carve-up (floats): xh | asrc | adst | asum | amax(int) | alphaE
  float* xh     = (float*)d_ws;                       // N*256
  float* asrc   = xh + (size_t)N * 256;               // N*4
  float* adst   = asrc + (size_t)N * 4;               // N*4
  float* asum   = adst + (size_t)N * 4;               // N*4
  int*   amax   = (int*)(asum + (size_t)N * 4);       // N*4
  float* alphaE = (float*)(amax + (size_t)N * 4);     // E*4

  const int* row = edge;
  const int* col = edge + E;

  const int outElems = N * 256;
  gat_init<<<(outElems + 255) / 256, 256, 0, stream>>>(out, bias, outElems, amax, asum, N * 4);

  gat_gemm<<<(N + 15) / 16, 256, 0, stream>>>(x, weight, xh, N);

  gat_alpha_node<<<(N * 4 + 255) / 256, 256, 0, stream>>>(xh, att_src, att_dst, asrc, adst, N);

  gat_edge_max<<<(E * 4 + 255) / 256, 256, 0, stream>>>(row, col, asrc, adst, alphaE, amax, E);

  gat_edge_exp<<<(E * 4 + 255) / 256, 256, 0, stream>>>(row, alphaE, amax, asum, E);

  gat_scatter<<<(E * 4 + 7) / 8, 256, 0, stream>>>(row, col, alphaE, asum, xh, out, E);
}